// Localization_57406532878662
// MI455X (gfx1250) — compile-verified
//
#include <hip/hip_runtime.h>
#include <math.h>

// ---------------------------------------------------------------------------
// Localization kernel for MI455X (gfx1250), fp32 WMMA path.
//   dist^2[b,f] = X2 @ S2^T + X @ W2^T + c[f]
//     X2 = x^2, S2 = sigma^2, W2 = -2*sigma^2*mu, c[f] = sum_d (sigma*mu)^2
//   out = softmax_F( sigmoid(T) * exp(-sqrt(max(dist^2,0))) )
// ---------------------------------------------------------------------------

typedef __attribute__((ext_vector_type(2))) float v2f;
typedef __attribute__((ext_vector_type(8))) float v8f;

#define B_DIM 4096
#define F_DIM 512
#define D_DIM 512

#define M_TILE 128
#define N_TILE 128
#define K_TILE 32
#define LDS_STRIDE 36   // pad: conflict-free b64 frag reads + 16B-aligned stores

__device__ __forceinline__ v8f wmma_f32(v2f a, v2f b, v8f c) {
    // V_WMMA_F32_16X16X4_F32 : D = A(16x4 f32) * B(4x16 f32) + C(16x16 f32)
    return __builtin_amdgcn_wmma_f32_16x16x4_f32(
        /*neg_a=*/false, a, /*neg_b=*/false, b,
        /*c_mod=*/(short)0, c, /*reuse_a=*/false, /*reuse_b=*/false);
}

// ---- Pass 0: c[f] = sum_d (sigma[f,d]*mu[f,d])^2 --------------------------
__global__ __launch_bounds__(256) void loc_prep_cf(const float* __restrict__ MU,
                                                   const float* __restrict__ SG,
                                                   float* __restrict__ CF) {
    __shared__ float red[256];
    const int f = blockIdx.x;
    const int tid = threadIdx.x;
    float s = 0.f;
    for (int d = tid; d < D_DIM; d += 256) {
        float t = SG[(size_t)f * D_DIM + d] * MU[(size_t)f * D_DIM + d];
        s += t * t;
    }
    red[tid] = s;
    __syncthreads();
    for (int stp = 128; stp > 0; stp >>= 1) {
        if (tid < stp) red[tid] += red[tid + stp];
        __syncthreads();
    }
    if (tid == 0) CF[f] = red[0];
}

// ---- Pass 1: WMMA GEMM + fused exp(-sqrt(.)) epilogue ---------------------
__global__ __launch_bounds__(256) void loc_gemm_wmma(const float* __restrict__ X,
                                                     const float* __restrict__ MU,
                                                     const float* __restrict__ SG,
                                                     const float* __restrict__ CF,
                                                     float* __restrict__ OUT) {
    __shared__ float Xs [M_TILE * LDS_STRIDE];
    __shared__ float S2s[N_TILE * LDS_STRIDE];
    __shared__ float W2s[N_TILE * LDS_STRIDE];

    const int tid  = threadIdx.x;
    const int bm   = blockIdx.x * M_TILE;   // 32 blocks
    const int bn   = blockIdx.y * N_TILE;   // 4 blocks

    const int wave = tid >> 5;              // 0..7 (wave32)
    const int lane = tid & 31;
    const int wm   = (wave & 3) * 32;       // wave M offset within block tile
    const int wn   = (wave >> 2) * 64;      // wave N offset within block tile

    v8f acc[2][4];
#pragma unroll
    for (int i = 0; i < 2; ++i)
#pragma unroll
        for (int j = 0; j < 4; ++j) acc[i][j] = (v8f)0.f;

    // cooperative loader coordinates: 256 threads, float4 each, 32 rows/pass
    const int lrow = tid >> 3;          // 0..31
    const int lcol = (tid & 7) * 4;     // 0,4,...,28

    for (int kt = 0; kt < D_DIM; kt += K_TILE) {
        // stage X tile (raw x; x^2 is formed in registers at frag time)
#pragma unroll
        for (int rr = 0; rr < M_TILE; rr += 32) {
            const int r = lrow + rr;
            const float4 xv =
                *(const float4*)(X + (size_t)(bm + r) * D_DIM + kt + lcol);
            *(float4*)&Xs[r * LDS_STRIDE + lcol] = xv;
        }
        // stage sigma/mu tiles, transformed: S2 = s^2, W2 = -2*s^2*mu
#pragma unroll
        for (int rr = 0; rr < N_TILE; rr += 32) {
            const int r = lrow + rr;
            const float4 sg =
                *(const float4*)(SG + (size_t)(bn + r) * D_DIM + kt + lcol);
            const float4 mu =
                *(const float4*)(MU + (size_t)(bn + r) * D_DIM + kt + lcol);
            float4 s2, w2;
            s2.x = sg.x * sg.x; s2.y = sg.y * sg.y;
            s2.z = sg.z * sg.z; s2.w = sg.w * sg.w;
            w2.x = -2.f * s2.x * mu.x; w2.y = -2.f * s2.y * mu.y;
            w2.z = -2.f * s2.z * mu.z; w2.w = -2.f * s2.w * mu.w;
            *(float4*)&S2s[r * LDS_STRIDE + lcol] = s2;
            *(float4*)&W2s[r * LDS_STRIDE + lcol] = w2;
        }
        // prefetch next X tile into cache hierarchy (global_prefetch_b8)
        if (kt + K_TILE < D_DIM) {
            __builtin_prefetch(X + (size_t)(bm + lrow) * D_DIM + kt + K_TILE + lcol, 0, 1);
        }
        __syncthreads();

        const int arow = lane & 15;
        const int khi  = (lane >> 4) << 1;   // 0 or 2
#pragma unroll
        for (int kk = 0; kk < K_TILE; kk += 4) {
            const int kb = kk + khi;
            // A fragments (two 16-row subtiles) + squared versions
            const v2f a0 = *(const v2f*)&Xs[(wm + arow)      * LDS_STRIDE + kb];
            const v2f a1 = *(const v2f*)&Xs[(wm + 16 + arow) * LDS_STRIDE + kb];
            const v2f a0q = a0 * a0;
            const v2f a1q = a1 * a1;
#pragma unroll
            for (int ns = 0; ns < 4; ++ns) {
                const int frow = wn + ns * 16 + arow;
                const v2f bs = *(const v2f*)&S2s[frow * LDS_STRIDE + kb];
                const v2f bw = *(const v2f*)&W2s[frow * LDS_STRIDE + kb];
                acc[0][ns] = wmma_f32(a0q, bs, acc[0][ns]);
                acc[0][ns] = wmma_f32(a0,  bw, acc[0][ns]);
                acc[1][ns] = wmma_f32(a1q, bs, acc[1][ns]);
                acc[1][ns] = wmma_f32(a1,  bw, acc[1][ns]);
            }
        }
        __syncthreads();
    }

    // epilogue: pre = acc + c[f];  loc = exp(-sqrt(max(pre,0)))
    const int nlo = lane & 15;
    const int mhi = (lane >> 4) * 8;
#pragma unroll
    for (int ms = 0; ms < 2; ++ms) {
#pragma unroll
        for (int ns = 0; ns < 4; ++ns) {
            const int col = bn + wn + ns * 16 + nlo;
            const float cf = CF[col];
#pragma unroll
            for (int v = 0; v < 8; ++v) {
                const int row = bm + wm + ms * 16 + mhi + v;
                const float pre  = acc[ms][ns][v] + cf;
                const float dist = sqrtf(fmaxf(pre, 0.f));
                OUT[(size_t)row * F_DIM + col] = __expf(-dist);
            }
        }
    }
}

// ---- Pass 2: in-place softmax over F with sigmoid(T) scaling --------------
__global__ __launch_bounds__(256) void loc_softmax(float* __restrict__ OUT,
                                                   const float* __restrict__ TEMP) {
    __shared__ float red[256];
    const int b   = blockIdx.x;
    const int tid = threadIdx.x;
    const float g = 1.f / (1.f + __expf(-TEMP[0]));

    float* row = OUT + (size_t)b * F_DIM;
    const float z0 = g * row[tid];
    const float z1 = g * row[tid + 256];

    red[tid] = fmaxf(z0, z1);
    __syncthreads();
    for (int s = 128; s > 0; s >>= 1) {
        if (tid < s) red[tid] = fmaxf(red[tid], red[tid + s]);
        __syncthreads();
    }
    const float mx = red[0];
    __syncthreads();

    const float e0 = __expf(z0 - mx);
    const float e1 = __expf(z1 - mx);
    red[tid] = e0 + e1;
    __syncthreads();
    for (int s = 128; s > 0; s >>= 1) {
        if (tid < s) red[tid] += red[tid + s];
        __syncthreads();
    }
    const float inv = 1.f / red[0];
    row[tid]       = e0 * inv;
    row[tid + 256] = e1 * inv;
}

// ---------------------------------------------------------------------------
extern "C" void kernel_launch(void* const* d_in, const int* in_sizes, int n_in,
                              void* d_out, int out_size, void* d_ws, size_t ws_size,
                              hipStream_t stream) {
    const float* X  = (const float*)d_in[0];   // inputs  [4096,512]
    const float* MU = (const float*)d_in[1];   // mu      [512,512]
    const float* SG = (const float*)d_in[2];   // sigma   [1,512,512]
    const float* T  = (const float*)d_in[3];   // temperature scalar
    float* OUT = (float*)d_out;                // [4096,512]
    float* CF  = (float*)d_ws;                 // 512 floats scratch

    loc_prep_cf<<<F_DIM, 256, 0, stream>>>(MU, SG, CF);

    dim3 grid(B_DIM / M_TILE, F_DIM / N_TILE);   // (32, 4)
    loc_gemm_wmma<<<grid, 256, 0, stream>>>(X, MU, SG, CF, OUT);

    loc_softmax<<<B_DIM, 256, 0, stream>>>(OUT, T);
}